// Attention_37538014167310
// MI455X (gfx1250) — compile-verified
//
#include <hip/hip_runtime.h>
#include <stdint.h>

// ---------------------------------------------------------------------------
// Local-window attention (B=2, N=2048, C=768, H=12, Dh=64, window=±7), fp32.
// Matrix math: V_WMMA_F32_16X16X4_F32 (CDNA5 fp32 WMMA).
// GEMM staging: double-buffered GLOBAL_LOAD_ASYNC_TO_LDS_B128 (ASYNCcnt),
// falling back to sync global->VGPR->LDS staging if the builtin is absent.
// ---------------------------------------------------------------------------

typedef float v2f __attribute__((ext_vector_type(2)));
typedef float v8f __attribute__((ext_vector_type(8)));
typedef int   v4i __attribute__((vector_size(4 * sizeof(int))));

#if defined(__has_builtin)
#if __has_builtin(__builtin_amdgcn_global_load_async_to_lds_b128)
#define HAVE_ASYNC_LDS 1
#endif
#endif
#ifndef HAVE_ASYNC_LDS
#define HAVE_ASYNC_LDS 0
#endif

#if HAVE_ASYNC_LDS
typedef __attribute__((address_space(1))) v4i* g128_t; // global 16B vector ptr
typedef __attribute__((address_space(3))) v4i* l128_t; // LDS    16B vector ptr
#endif

static __device__ __forceinline__ v8f wmma_f32(v2f a, v2f b, v8f c) {
  // D(16x16,f32) = A(16x4,f32) * B(4x16,f32) + C
  return __builtin_amdgcn_wmma_f32_16x16x4_f32(
      false, a, false, b, (short)0, c, false, false);
}

// Wait for all of this wave's outstanding async LDS loads (ASYNCcnt == 0).
static __device__ __forceinline__ void async_wait0() {
#if HAVE_ASYNC_LDS
#if __has_builtin(__builtin_amdgcn_s_wait_asynccnt)
  __builtin_amdgcn_s_wait_asynccnt(0);
#else
  asm volatile("s_wait_asynccnt 0x0" ::: "memory");
#endif
#endif
}

// Copy 16 bytes global -> LDS. Async (no VGPR round-trip) when available.
static __device__ __forceinline__ void lds_copy16(const float* g, float* l) {
#if HAVE_ASYNC_LDS
  // AS1 pointer: same bit pattern as the generic global pointer.
  // AS3 pointer: low 32 bits of a generic LDS pointer are the LDS byte
  // offset (ISA LDS aperture truncation rule).
  __builtin_amdgcn_global_load_async_to_lds_b128(
      (g128_t)(uintptr_t)g, (l128_t)(uint32_t)(uintptr_t)l, 0, 0);
#else
  *(float4*)l = *(const float4*)g;
#endif
}

// ---------------------------------------------------------------------------
// Generic  Y[M,N] = X[M,K] @ W[N,K]^T (+ bias[N])   -- fp32 WMMA GEMM.
// Block: 256 threads = 8 waves. Block tile 128x128, wave tile 32x64.
// Double-buffered LDS staging (stride-20 pad: 16B-aligned b128 stores,
// conflict-free b64 fragment loads). Requires M%128==0, N%128==0, K%16==0.
// ---------------------------------------------------------------------------
template <bool HAS_BIAS>
__global__ __launch_bounds__(256) void gemm_tn_wmma(
    const float* __restrict__ X, const float* __restrict__ W,
    const float* __restrict__ bias, float* __restrict__ Y,
    int M, int N, int K) {
  constexpr int BM = 128, BN = 128, BK = 16, LDT = 20;
  __shared__ float XsBuf[2 * BM * LDT];
  __shared__ float WsBuf[2 * BN * LDT];

  const int tid  = threadIdx.x;
  const int lane = tid & 31;
  const int wave = tid >> 5;
  const int wm   = (wave & 3) * 32;  // wave row base within block tile
  const int wn   = (wave >> 2) * 64; // wave col base within block tile
  const int Mb   = blockIdx.y * BM;
  const int Nb   = blockIdx.x * BN;

  const int half = (lane < 16) ? 0 : 2; // K sub-pair owned by this lane half
  const int l16  = lane & 15;

  v8f acc[2][4] = {};

  // stage one BK-wide K chunk of X and W into LDS buffer `buf`
  auto stage = [&](int kc, int buf) {
    float* xs = &XsBuf[buf * BM * LDT];
    float* ws = &WsBuf[buf * BN * LDT];
#pragma unroll
    for (int p = 0; p < 2; ++p) {
      int id  = tid + p * 256;      // 0..511
      int row = id >> 2;            // 0..127
      int seg = id & 3;             // float4 segment within the 16-wide chunk
      lds_copy16(X + (size_t)(Mb + row) * K + kc + seg * 4,
                 &xs[row * LDT + seg * 4]);
      lds_copy16(W + (size_t)(Nb + row) * K + kc + seg * 4,
                 &ws[row * LDT + seg * 4]);
    }
  };

  stage(0, 0);
  int cur = 0;

  for (int kc = 0; kc < K; kc += BK) {
    async_wait0();      // this wave's async loads for buffer `cur` done
    __syncthreads();    // ... and every other wave's too

    if (kc + BK < K) stage(kc + BK, cur ^ 1);  // overlap next stage w/ compute

    const float* xs = &XsBuf[cur * BM * LDT];
    const float* ws = &WsBuf[cur * BN * LDT];
#pragma unroll
    for (int k0 = 0; k0 < BK; k0 += 4) {
      v2f aF[2], bF[4];
#pragma unroll
      for (int ms = 0; ms < 2; ++ms)
        aF[ms] = *(const v2f*)(&xs[(wm + ms * 16 + l16) * LDT + k0 + half]);
#pragma unroll
      for (int ns = 0; ns < 4; ++ns)
        bF[ns] = *(const v2f*)(&ws[(wn + ns * 16 + l16) * LDT + k0 + half]);
#pragma unroll
      for (int ms = 0; ms < 2; ++ms)
#pragma unroll
        for (int ns = 0; ns < 4; ++ns)
          acc[ms][ns] = wmma_f32(aF[ms], bF[ns], acc[ms][ns]);
    }
    __syncthreads();    // all waves done reading `cur` before it is restaged
    cur ^= 1;
  }

  // --- epilogue: C/D layout = lane%16 -> col, VGPR r -> row r (+8 if lane>=16)
  const int mrow = (lane < 16) ? 0 : 8;
#pragma unroll
  for (int ms = 0; ms < 2; ++ms)
#pragma unroll
    for (int ns = 0; ns < 4; ++ns) {
      int n = Nb + wn + ns * 16 + l16;
      float bv = HAS_BIAS ? bias[n] : 0.0f;
#pragma unroll
      for (int r = 0; r < 8; ++r) {
        int m = Mb + wm + ms * 16 + r + mrow;
        Y[(size_t)m * N + n] = acc[ms][ns][r] + bv;
      }
    }
}

// ---------------------------------------------------------------------------
// Local-window attention. One wave handles one 16-query tile of one (b,h).
// Key band = 32 keys at jbase = i0-8 (covers |i-j|<=7 for all 16 queries).
// qkv layout: [B, N, 3C] fp32 (C = H*Dh).  out: [B, N, C].
// ---------------------------------------------------------------------------
__global__ __launch_bounds__(128) void local_attn_wmma(
    const float* __restrict__ qkv, float* __restrict__ out,
    int B, int N, int C, int H) {
  constexpr int Dh  = 64;
  constexpr int WIN = 7;

  const int lane = threadIdx.x & 31;
  const int wave = threadIdx.x >> 5;
  const int tile = blockIdx.x * 4 + wave;
  const int ntq  = N / 16;

  const int qt = tile % ntq;
  const int h  = (tile / ntq) % H;
  const int b  = tile / (ntq * H);
  const int i0 = qt * 16;
  const int jbase = i0 - 8;

  const float scale = 0.125f; // 1/sqrt(64)
  const size_t tok = (size_t)3 * C; // floats per token in qkv buffer

  const float* qbase = qkv + (size_t)b * N * tok + (size_t)h * Dh;
  const float* kbase = qbase + C;
  const float* vbase = qbase + 2 * C;

  __shared__ float S_all[4][16][34];       // scores / probabilities
  __shared__ float V_all[4][32][Dh + 4];   // key-band V, 16B-aligned rows
  float (*S)[34]      = S_all[wave];
  float (*Vl)[Dh + 4] = V_all[wave];

  const int half = (lane < 16) ? 0 : 2;
  const int l16  = lane & 15;
  const int mrow = (lane < 16) ? 0 : 8;

  // --- stage V band (32 keys x 64) into LDS, one key row per lane ----------
  {
    int key = jbase + lane;
    key = min(max(key, 0), N - 1);
    const float* vrow = vbase + (size_t)key * tok;
#pragma unroll
    for (int d = 0; d < Dh; d += 4)
      *(float4*)(&Vl[lane][d]) = *(const float4*)(vrow + d);
  }

  // --- S(16x32) = Q(16x64) @ K_band(32x64)^T, straight from global ---------
  v8f accS[2] = {};
  {
    const float* qrow = qbase + (size_t)(i0 + l16) * tok;
    int j0 = min(max(jbase + l16, 0), N - 1);
    int j1 = min(max(jbase + 16 + l16, 0), N - 1);
    const float* krow0 = kbase + (size_t)j0 * tok;
    const float* krow1 = kbase + (size_t)j1 * tok;
#pragma unroll
    for (int d0 = 0; d0 < Dh; d0 += 4) {
      v2f a  = *(const v2f*)(qrow + d0 + half);
      v2f b0 = *(const v2f*)(krow0 + d0 + half);
      v2f b1 = *(const v2f*)(krow1 + d0 + half);
      accS[0] = wmma_f32(a, b0, accS[0]);
      accS[1] = wmma_f32(a, b1, accS[1]);
    }
  }

  // --- scale + local-window mask, dump to LDS ------------------------------
#pragma unroll
  for (int t = 0; t < 2; ++t) {
    int j = jbase + t * 16 + l16;
#pragma unroll
    for (int r = 0; r < 8; ++r) {
      int m = r + mrow;
      int i = i0 + m;
      int dj = i - j;
      bool ok = (j >= 0) && (j < N) && (dj <= WIN) && (dj >= -WIN);
      S[m][t * 16 + l16] = ok ? accS[t][r] * scale : -__builtin_inff();
    }
  }
  __syncthreads();

  // --- row softmax over the 32-wide band (lanes 0..15, one row each) -------
  if (lane < 16) {
    float mx = -__builtin_inff();
#pragma unroll
    for (int c = 0; c < 32; ++c) mx = fmaxf(mx, S[lane][c]);
    float sum = 0.0f;
#pragma unroll
    for (int c = 0; c < 32; ++c) {
      float e = __expf(S[lane][c] - mx);
      S[lane][c] = e;
      sum += e;
    }
    float inv = 1.0f / sum;
#pragma unroll
    for (int c = 0; c < 32; ++c) S[lane][c] *= inv;
  }
  __syncthreads();

  // --- O(16x64) = P(16x32) @ V_band(32x64) ---------------------------------
  v8f accO[4] = {};
#pragma unroll
  for (int k0 = 0; k0 < 32; k0 += 4) {
    v2f a = *(const v2f*)(&S[l16][k0 + half]);
#pragma unroll
    for (int ns = 0; ns < 4; ++ns) {
      v2f bf;
      bf.x = Vl[k0 + half][ns * 16 + l16];
      bf.y = Vl[k0 + half + 1][ns * 16 + l16];
      accO[ns] = wmma_f32(a, bf, accO[ns]);
    }
  }

  // --- write O -------------------------------------------------------------
  float* obase = out + ((size_t)b * N) * C + (size_t)h * Dh;
#pragma unroll
  for (int ns = 0; ns < 4; ++ns)
#pragma unroll
    for (int r = 0; r < 8; ++r) {
      int m = r + mrow;
      obase[(size_t)(i0 + m) * C + ns * 16 + l16] = accO[ns][r];
    }
}

// ---------------------------------------------------------------------------
extern "C" void kernel_launch(void* const* d_in, const int* in_sizes, int n_in,
                              void* d_out, int out_size, void* d_ws, size_t ws_size,
                              hipStream_t stream) {
  (void)in_sizes; (void)n_in; (void)out_size; (void)ws_size;
  constexpr int B = 2, N = 2048, C = 768, H = 12;

  const float* x      = (const float*)d_in[0];
  const float* w_qkv  = (const float*)d_in[1];
  const float* w_proj = (const float*)d_in[2];
  const float* b_proj = (const float*)d_in[3];
  float* out = (float*)d_out;

  float* qkv_ws  = (float*)d_ws;                       // [B,N,3C]  37.7 MB
  float* attn_ws = qkv_ws + (size_t)B * N * 3 * C;     // [B,N,C]   12.6 MB

  // 1) QKV projection: [4096,768] @ [2304,768]^T -> [4096,2304]
  {
    dim3 grid((3 * C) / 128, (B * N) / 128);
    gemm_tn_wmma<false><<<grid, 256, 0, stream>>>(
        x, w_qkv, nullptr, qkv_ws, B * N, 3 * C, C);
  }

  // 2) local-window attention: 3072 query tiles, 4 waves/block
  {
    int tiles = B * H * (N / 16);
    local_attn_wmma<<<dim3(tiles / 4), 128, 0, stream>>>(
        qkv_ws, attn_ws, B, N, C, H);
  }

  // 3) output projection (+bias): [4096,768] @ [768,768]^T -> [4096,768]
  {
    dim3 grid(C / 128, (B * N) / 128);
    gemm_tn_wmma<true><<<grid, 256, 0, stream>>>(
        attn_ws, w_proj, b_proj, out, B * N, C, C);
  }
}